// MoESwiGLUFFN_9380208575128
// MI455X (gfx1250) — compile-verified
//
#include <hip/hip_runtime.h>
#include <math.h>

#define E        512
#define HDIM     1365
#define NE       8
#define LN_EPS   1e-5f

#define TILE_T   16
#define WAVES    8
#define BLK      256
#define HCHUNKS  86        // ceil(1365/16)
#define XS_PITCH 516       // 512 + 4 pad: row pitch 2064 B (16B-aligned), bank stride 4 -> conflict-free
#define HS_PITCH 20        // 16 + 4 pad: bank stride 20 -> conflict-free

typedef __attribute__((ext_vector_type(2))) float v2f;
typedef __attribute__((ext_vector_type(8))) float v8f;

__device__ __forceinline__ v8f wmma_f32(v2f a, v2f b, v8f c) {
  // V_WMMA_F32_16X16X4_F32 : D(16x16,f32) = A(16x4,f32) * B(4x16,f32) + C
  return __builtin_amdgcn_wmma_f32_16x16x4_f32(false, a, false, b, (short)0, c,
                                               false, false);
}

// Low 32 bits of a generic pointer into LDS == byte offset from the wave's LDS base
// this is what the async-to-LDS VDST operand wants.
__device__ __forceinline__ unsigned lds_off(const void* p) {
  return (unsigned)(uintptr_t)p;
}

// CDNA5 async copy: memory -> LDS, 16 bytes per lane, tracked by ASYNCcnt (no VGPR data path).
__device__ __forceinline__ void async_load_b128(unsigned lds_addr, const void* gaddr) {
  asm volatile("global_load_async_to_lds_b128 %0, %1, off"
               :: "v"(lds_addr), "v"((unsigned long long)(uintptr_t)gaddr)
               : "memory");
}
__device__ __forceinline__ void wait_asynccnt0() {
  asm volatile("s_wait_asynccnt 0x0" ::: "memory");
}

// ---------------------------------------------------------------- k0: init
__global__ void k0_init(float* __restrict__ outacc, int* __restrict__ meta, int total) {
  int i = blockIdx.x * blockDim.x + threadIdx.x;
  if (i < total) outacc[i] = 0.f;
  if (i < 24)    meta[i]   = 0;     // count[8], offset[8], cursor[8]
}

// ---------------------------------------------------------------- k1: router (one wave32 per token)
__global__ __launch_bounds__(BLK) void k1_router(
    const float* __restrict__ x, const float* __restrict__ rw,
    int* __restrict__ top_idx, float* __restrict__ top_w,
    int* __restrict__ meta, int n_tok)
{
  int wave = threadIdx.x >> 5;
  int lane = threadIdx.x & 31;
  int t = blockIdx.x * (BLK / 32) + wave;
  if (t >= n_tok) return;

  float acc[NE];
  #pragma unroll
  for (int e = 0; e < NE; ++e) acc[e] = 0.f;

  const float* xt = x + (size_t)t * E;
  for (int k = lane; k < E; k += 32) {
    float xv = xt[k];
    const float* r = rw + (size_t)k * NE;
    #pragma unroll
    for (int e = 0; e < NE; ++e) acc[e] += xv * r[e];
  }
  #pragma unroll
  for (int off = 16; off; off >>= 1) {
    #pragma unroll
    for (int e = 0; e < NE; ++e) acc[e] += __shfl_xor(acc[e], off, 32);
  }

  if (lane == 0) {
    int i0 = 0; float v0 = acc[0];
    #pragma unroll
    for (int e = 1; e < NE; ++e) if (acc[e] > v0) { v0 = acc[e]; i0 = e; }
    int i1 = (i0 == 0) ? 1 : 0; float v1 = acc[i1];
    #pragma unroll
    for (int e = 0; e < NE; ++e)
      if (e != i0 && acc[e] > v1) { v1 = acc[e]; i1 = e; }
    float e1 = __expf(v1 - v0);           // v1 <= v0, stable 2-way softmax
    float w0 = 1.f / (1.f + e1);
    float w1 = e1 * w0;
    top_idx[t * 2 + 0] = i0;  top_idx[t * 2 + 1] = i1;
    top_w [t * 2 + 0] = w0;  top_w [t * 2 + 1] = w1;
    atomicAdd(&meta[i0], 1);
    atomicAdd(&meta[i1], 1);
  }
}

// ---------------------------------------------------------------- k2: 8-wide prefix sum
__global__ void k2_prefix(int* __restrict__ meta) {
  if (threadIdx.x == 0) {
    int s = 0;
    for (int e = 0; e < NE; ++e) { meta[8 + e] = s; meta[16 + e] = s; s += meta[e]; }
  }
}

// ---------------------------------------------------------------- k3: scatter tokens into expert lists
__global__ void k3_scatter(const int* __restrict__ top_idx, const float* __restrict__ top_w,
                           int* __restrict__ meta, int* __restrict__ tok_list,
                           float* __restrict__ w_list, int n_tok)
{
  int t = blockIdx.x * blockDim.x + threadIdx.x;
  if (t >= n_tok) return;
  #pragma unroll
  for (int k = 0; k < 2; ++k) {
    int e = top_idx[t * 2 + k];
    int p = atomicAdd(&meta[16 + e], 1);
    tok_list[p] = t;
    w_list[p]   = top_w[t * 2 + k];
  }
}

// ---------------------------------------------------------------- k4: fused expert SwiGLU FFN
__global__ __launch_bounds__(BLK) void k4_expert(
    const float* __restrict__ x,
    const float* __restrict__ Wv, const float* __restrict__ bv,
    const float* __restrict__ Wg, const float* __restrict__ bg,
    const float* __restrict__ Wo, const float* __restrict__ bo,
    const int* __restrict__ meta, const int* __restrict__ tok_list,
    const float* __restrict__ w_list, float* __restrict__ outacc)
{
  __shared__ float xs[TILE_T][XS_PITCH];
  __shared__ float hstage[WAVES][TILE_T][HS_PITCH];
  __shared__ int   tok[TILE_T];
  __shared__ float wgt[TILE_T];

  const int e      = blockIdx.y;
  const int cnt    = meta[e];
  const int base   = meta[8 + e];
  const int tstart = blockIdx.x * TILE_T;
  if (tstart >= cnt) return;   // block-uniform early exit

  const int tid  = threadIdx.x;
  const int wave = tid >> 5;
  const int lane = tid & 31;
  const int half = lane >> 4;   // lane half select (K split in fp32 WMMA layouts)
  const int l16  = lane & 15;

  if (tid < TILE_T) {
    int slot   = tstart + tid;
    bool valid = slot < cnt;
    tok[tid] = tok_list[valid ? (base + slot) : base];
    wgt[tid] = valid ? w_list[base + slot] : 0.f;   // padded rows contribute 0
  }
  __syncthreads();
  // Gathered x tile: async memory->LDS (B128, ASYNCcnt), no VGPR staging.
  for (int i = tid; i < TILE_T * (E / 4); i += BLK) {
    int r = i >> 7, c4 = i & 127;                   // row, 16B chunk within row
    async_load_b128(lds_off(&xs[r][c4 * 4]),
                    x + (size_t)tok[r] * E + c4 * 4);
  }
  wait_asynccnt0();     // each wave drains its own async queue...
  __syncthreads();      // ...then the barrier publishes the whole tile

  const float* Wve = Wv + (size_t)e * E * HDIM;
  const float* Wge = Wg + (size_t)e * E * HDIM;
  const float* Woe = Wo + (size_t)e * HDIM * E;
  const float* bve = bv + (size_t)e * HDIM;
  const float* bge = bg + (size_t)e * HDIM;

  v8f acc0 = {}, acc1 = {}, acc2 = {}, acc3 = {};   // persistent 16x64 out slice per wave

  for (int cb = 0; cb < HCHUNKS; cb += WAVES) {
    const int chunk = cb + wave;
    // ---- phase 1: each wave builds one 16x16 SwiGLU h tile (wave-uniform branch) ----
    if (chunk < HCHUNKS) {
      v8f hv = {}, hg = {};
      const int jbase = chunk * 16;
      int jc = jbase + l16; jc = (jc > HDIM - 1) ? (HDIM - 1) : jc;   // branchless clamp
      #pragma unroll 4
      for (int kb = 0; kb < E / 4; ++kb) {
        const int k0 = kb * 4 + half * 2;
        v2f a;  a.x = xs[l16][k0];  a.y = xs[l16][k0 + 1];
        v2f bA, bB;
        bA.x = Wve[(size_t)k0 * HDIM + jc];
        bA.y = Wve[(size_t)(k0 + 1) * HDIM + jc];
        bB.x = Wge[(size_t)k0 * HDIM + jc];
        bB.y = Wge[(size_t)(k0 + 1) * HDIM + jc];
        hv = wmma_f32(a, bA, hv);
        hg = wmma_f32(a, bB, hg);
      }
      const float bvb = bve[jc], bgb = bge[jc];
      const bool colok = (jbase + l16) < HDIM;
      #pragma unroll
      for (int v = 0; v < 8; ++v) {
        int m = v + half * 8;
        float hvv = hv[v] + bvb;
        float hgg = hg[v] + bgb;
        // silu(hv)*hg with fast 1-ulp v_rcp_f32 instead of the IEEE divide expansion
        float act = hvv * __builtin_amdgcn_rcpf(1.f + __expf(-hvv)) * hgg;
        hstage[wave][m][l16] = colok ? act : 0.f;
      }
    } else {
      #pragma unroll
      for (int v = 0; v < 8; ++v) hstage[wave][v + half * 8][l16] = 0.f;
    }
    __syncthreads();
    // ---- phase 2: every wave folds the 8 staged h tiles into its 64 output columns ----
    for (int j2 = 0; j2 < WAVES; ++j2) {
      if (cb + j2 < HCHUNKS) {            // block-uniform
        const int kbase = (cb + j2) * 16;
        #pragma unroll
        for (int ks = 0; ks < 4; ++ks) {
          const int kl = ks * 4 + half * 2;
          v2f a;  a.x = hstage[j2][l16][kl];  a.y = hstage[j2][l16][kl + 1];
          int k  = kbase + kl;
          int kc  = (k     > HDIM - 1) ? (HDIM - 1) : k;       // rows >= HDIM hit h==0
          int kc1 = (k + 1 > HDIM - 1) ? (HDIM - 1) : (k + 1);
          const int colbase = wave * 64 + l16;
          v2f b;
          b.x = Woe[(size_t)kc * E + colbase +  0]; b.y = Woe[(size_t)kc1 * E + colbase +  0];
          acc0 = wmma_f32(a, b, acc0);
          b.x = Woe[(size_t)kc * E + colbase + 16]; b.y = Woe[(size_t)kc1 * E + colbase + 16];
          acc1 = wmma_f32(a, b, acc1);
          b.x = Woe[(size_t)kc * E + colbase + 32]; b.y = Woe[(size_t)kc1 * E + colbase + 32];
          acc2 = wmma_f32(a, b, acc2);
          b.x = Woe[(size_t)kc * E + colbase + 48]; b.y = Woe[(size_t)kc1 * E + colbase + 48];
          acc3 = wmma_f32(a, b, acc3);
        }
      }
    }
    __syncthreads();
  }

  // ---- combine-weight scale + bias, scatter into accumulator (top-2 overlap -> atomics) ----
  const float* boe = bo + (size_t)e * E;
  #pragma unroll
  for (int v = 0; v < 8; ++v) {
    int m = v + half * 8;
    float w = wgt[m];
    size_t row = (size_t)tok[m] * E;
    int col = wave * 64 + l16;
    atomicAdd(&outacc[row + col +  0], w * (acc0[v] + boe[col +  0]));
    atomicAdd(&outacc[row + col + 16], w * (acc1[v] + boe[col + 16]));
    atomicAdd(&outacc[row + col + 32], w * (acc2[v] + boe[col + 32]));
    atomicAdd(&outacc[row + col + 48], w * (acc3[v] + boe[col + 48]));
  }
}

// ---------------------------------------------------------------- k5: residual + LayerNorm (in-place on out)
__global__ __launch_bounds__(256) void k5_ln(
    const float* __restrict__ x, const float* __restrict__ gamma,
    const float* __restrict__ beta, float* __restrict__ out)
{
  __shared__ float red[256];
  const int t = blockIdx.x;
  const int c0 = threadIdx.x, c1 = threadIdx.x + 256;
  const size_t b = (size_t)t * E;
  float y0 = x[b + c0] + out[b + c0];
  float y1 = x[b + c1] + out[b + c1];

  red[threadIdx.x] = y0 + y1;
  __syncthreads();
  for (int off = 128; off; off >>= 1) {
    if (threadIdx.x < off) red[threadIdx.x] += red[threadIdx.x + off];
    __syncthreads();
  }
  float mu = red[0] / (float)E;
  __syncthreads();

  float d0 = y0 - mu, d1 = y1 - mu;
  red[threadIdx.x] = d0 * d0 + d1 * d1;
  __syncthreads();
  for (int off = 128; off; off >>= 1) {
    if (threadIdx.x < off) red[threadIdx.x] += red[threadIdx.x + off];
    __syncthreads();
  }
  float inv = rsqrtf(red[0] / (float)E + LN_EPS);

  out[b + c0] = gamma[c0] * d0 * inv + beta[c0];
  out[b + c1] = gamma[c1] * d1 * inv + beta[c1];
}

// ---------------------------------------------------------------- launcher
extern "C" void kernel_launch(void* const* d_in, const int* in_sizes, int n_in,
                              void* d_out, int out_size, void* d_ws, size_t ws_size,
                              hipStream_t stream)
{
  const float* x     = (const float*)d_in[0];
  const float* rw    = (const float*)d_in[1];
  const float* Wv    = (const float*)d_in[2];
  const float* bv    = (const float*)d_in[3];
  const float* Wg    = (const float*)d_in[4];
  const float* bg    = (const float*)d_in[5];
  const float* Wo    = (const float*)d_in[6];
  const float* bo    = (const float*)d_in[7];
  const float* gamma = (const float*)d_in[8];
  const float* beta  = (const float*)d_in[9];
  float* out = (float*)d_out;

  const int n_tok = in_sizes[0] / E;   // 8192

  // workspace layout (~256 KB): top_idx | top_w | meta(count/offset/cursor) | tok_list | w_list
  char*  ws       = (char*)d_ws;
  int*   top_idx  = (int*)(ws);
  float* top_w    = (float*)(ws + (size_t)n_tok * 2 * 4);
  int*   meta     = (int*)(ws + (size_t)n_tok * 4 * 4);
  int*   tok_list = (int*)(ws + (size_t)n_tok * 4 * 4 + 128);
  float* w_list   = (float*)(ws + (size_t)n_tok * 4 * 4 + 128 + (size_t)n_tok * 2 * 4);

  const int total = n_tok * E;
  k0_init   <<<(total + 255) / 256, 256, 0, stream>>>(out, meta, total);
  k1_router <<<(n_tok + 7) / 8, BLK, 0, stream>>>(x, rw, top_idx, top_w, meta, n_tok);
  k2_prefix <<<1, 32, 0, stream>>>(meta);
  k3_scatter<<<(n_tok + 255) / 256, 256, 0, stream>>>(top_idx, top_w, meta, tok_list, w_list, n_tok);
  dim3 g4((n_tok + TILE_T - 1) / TILE_T, NE);   // worst case: one expert owns every token
  k4_expert <<<g4, BLK, 0, stream>>>(x, Wv, bv, Wg, bg, Wo, bo, meta, tok_list, w_list, out);
  k5_ln     <<<n_tok, 256, 0, stream>>>(x, gamma, beta, out);
}